// C_Convolution_25082609009494
// MI455X (gfx1250) — compile-verified
//
#include <hip/hip_runtime.h>

typedef float v2f __attribute__((ext_vector_type(2)));
typedef float v8f __attribute__((ext_vector_type(8)));

#define CIN   4
#define COUT  4
#define LX    16
#define LY    16
#define LZ    16
#define LT    32
#define NCOMP 384              // LT * spin*color = 32*12 components per (cin,x,y,z) pencil
#define PENCIL_F (NCOMP * 2)   // floats per (pencil, cin)
#define WFLOATS (CIN * COUT * 81 * 2)   // 2592
#define UBUF_F  (3 * CIN * NCOMP * 2)   // 9216 floats per stage buffer

// LDS byte offset of a generic pointer to LDS: flat addr = {SHARED_BASE, offset}, so low 32 bits.
__device__ __forceinline__ unsigned int lds_off(const void* p) {
  return (unsigned int)(unsigned long long)p;
}

__device__ __forceinline__ void async_b128(unsigned int lds_addr, unsigned long long gaddr) {
  asm volatile("global_load_async_to_lds_b128 %0, %1, off"
               :: "v"(lds_addr), "v"(gaddr) : "memory");
}

// out[o,x,y,z,t,s,c] = b[o] + sum_{a,by,cz,dt,i} W[i,o,a,by,cz,dt] * U[i, x+a-1, y+by-1, z+cz-1, t+dt-1, s, c]
__global__ __launch_bounds__(128) void conv4d_cplx_wmma(
    const float* __restrict__ U,   // (4,16,16,16,32,4,3) complex64 -> interleaved floats
    const float* __restrict__ W,   // (4,4,3,3,3,3) complex64
    const float* __restrict__ B,   // (4,) complex64
    float* __restrict__ out)       // (4,16,16,16,32,4,3) complex64
{
  extern __shared__ float smem[];
  float* sW  = smem;               // 2592 floats (10.1 KB)
  float* sU0 = smem + WFLOATS;     // 9216 floats (36 KB)
  float* sU1 = sU0 + UBUF_F;       // 9216 floats (36 KB)

  const int tid    = threadIdx.x;
  const int lane   = tid & 31;
  const int wave   = tid >> 5;
  const int lanelo = lane & 15;
  const int lanehi = lane >> 4;    // 0 or 1 (half-wave)
  const int x = blockIdx.x, y = blockIdx.y, z = blockIdx.z;

  // ---- async DMA issue for one halo stage: 3 x-neighbor pencils, all C_in (2304 b128) ----
  auto issue_stage = [&](int s, float* buf) {
    const int cz = s / 3, by = s - 3 * (s / 3);
    const int zs = (z + cz + LZ - 1) & (LZ - 1);
    const int ys = (y + by + LY - 1) & (LY - 1);
    for (int idx = tid; idx < 3 * CIN * (PENCIL_F / 4); idx += 128) {
      const int a  = idx / (CIN * (PENCIL_F / 4));
      const int r  = idx - a * (CIN * (PENCIL_F / 4));
      const int i  = r / (PENCIL_F / 4);
      const int ch = r - i * (PENCIL_F / 4);
      const int xs = (x + a + LX - 1) & (LX - 1);
      const long gbase = (long)(((i * LX + xs) * LY + ys) * LZ + zs) * NCOMP;  // complex units
      const unsigned long long ga =
          (unsigned long long)(const void*)(U + 2 * gbase) + (unsigned long long)ch * 16ull;
      const unsigned int la = lds_off(&buf[((a * CIN + i) * (PENCIL_F / 4) + ch) * 4]);
      async_b128(la, ga);
    }
  };

  // ---- kick off: whole W (648 b128) + halo stage 0, all via the async-to-LDS path ----
  for (int idx = tid; idx < WFLOATS / 4; idx += 128) {
    async_b128(lds_off(&sW[idx * 4]),
               (unsigned long long)(const void*)W + (unsigned long long)idx * 16ull);
  }
  issue_stage(0, sU0);

  // 6 output tiles (16x16 f32) per wave, live in VGPRs across all 81 offsets
  v8f acc[6];
#pragma unroll
  for (int t = 0; t < 6; ++t) acc[t] = (v8f){};

  // B-matrix column -> (output channel, re/im)
  const int  col   = lanelo;
  const int  oo    = (col >> 1) & 3;
  const int  pp    = col & 1;
  const bool colok = (col < 8);

  const float2* sW2 = (const float2*)sW;

  for (int s = 0; s < 9; ++s) {
    float* cur = (s & 1) ? sU1 : sU0;
    float* nxt = (s & 1) ? sU0 : sU1;

    // consume barrier: my DMA into `cur` done, then all waves' done (and all stage s-1 reads done)
    asm volatile("s_wait_asynccnt 0x0" ::: "memory");
    __syncthreads();

    // overlap: DMA next stage into the other buffer while we compute on `cur`
    if (s + 1 < 9) issue_stage(s + 1, nxt);

    const int cz = s / 3, by = s - 3 * (s / 3);
    const float2* sU2 = (const float2*)cur;

    for (int a = 0; a < 3; ++a) {
      for (int dt = 0; dt < 3; ++dt) {
        // B fragments (W complex embedding), k-step0: cin {0,1}, k-step1: cin {2,3}
        // Lane layout mirrors documented A layout: v0 = rows {0|2}, v1 = rows {1|3}.
        v2f b0, b1;
        {
          const int i0 = lanehi;
          const int w0 = ((((i0 * COUT + oo) * 3 + a) * 3 + by) * 3 + cz) * 3 + dt;
          const float2 wc = sW2[w0];
          b0.x = colok ? (pp ? wc.y :  wc.x) : 0.f;   // row (i,re): [Wre | Wim]
          b0.y = colok ? (pp ? wc.x : -wc.y) : 0.f;   // row (i,im): [-Wim | Wre]
          const int i1 = 2 + lanehi;
          const int w1 = ((((i1 * COUT + oo) * 3 + a) * 3 + by) * 3 + cz) * 3 + dt;
          const float2 wd = sW2[w1];
          b1.x = colok ? (pp ? wd.y :  wd.x) : 0.f;
          b1.y = colok ? (pp ? wd.x : -wd.y) : 0.f;
        }
        const int shift = 12 * (dt - 1);  // t-shift folded into component index

#pragma unroll
        for (int tt = 0; tt < 6; ++tt) {
          const int nb = (wave * 6 + tt) * 16;
          int m = nb + lanelo + shift;
          m += (m < 0)      ? NCOMP : 0;   // branchless periodic wrap
          m -= (m >= NCOMP) ? NCOMP : 0;

          // A fragments: lane holds (re,im) of U[cin, component m] -> one b64 LDS load each
          const float2 ua0 = sU2[(a * CIN +     lanehi) * NCOMP + m];
          const float2 ua1 = sU2[(a * CIN + 2 + lanehi) * NCOMP + m];
          v2f av0; av0.x = ua0.x; av0.y = ua0.y;
          v2f av1; av1.x = ua1.x; av1.y = ua1.y;

          acc[tt] = __builtin_amdgcn_wmma_f32_16x16x4_f32(
              false, av0, false, b0, (short)0, acc[tt], false, false);
          acc[tt] = __builtin_amdgcn_wmma_f32_16x16x4_f32(
              false, av1, false, b1, (short)0, acc[tt], false, false);
        }
      }
    }
  }

  // ---- store: D[j] holds row M=j (lanes 0-15) / M=j+8 (lanes 16-31), N=col ----
  if (colok) {
    const float bias = B[oo * 2 + pp];
#pragma unroll
    for (int tt = 0; tt < 6; ++tt) {
      const int nb = (wave * 6 + tt) * 16;
#pragma unroll
      for (int j = 0; j < 8; ++j) {
        const int  n    = nb + j + (lanehi ? 8 : 0);           // t*12 + s*3 + c
        const long cidx = ((long)(((oo * LX + x) * LY + y) * LZ + z)) * NCOMP + n;
        out[cidx * 2 + pp] = acc[tt][j] + bias;
      }
    }
  }
}

extern "C" void kernel_launch(void* const* d_in, const int* in_sizes, int n_in,
                              void* d_out, int out_size, void* d_ws, size_t ws_size,
                              hipStream_t stream) {
  const float* U = (const float*)d_in[0];
  const float* W = (const float*)d_in[1];
  const float* b = (const float*)d_in[2];
  float* out = (float*)d_out;
  dim3 grid(LX, LY, LZ);
  dim3 block(128);
  const size_t shmem = (size_t)(WFLOATS + 2 * UBUF_F) * sizeof(float);  // 84,096 B
  hipLaunchKernelGGL(conv4d_cplx_wmma, grid, block, shmem, stream, U, W, b, out);
}